// GCN_76149770158504
// MI455X (gfx1250) — compile-verified
//
#include <hip/hip_runtime.h>
#include <hip/hip_bf16.h>

typedef __attribute__((ext_vector_type(2))) float v2f;
typedef __attribute__((ext_vector_type(8))) float v8f;

// ---------------------------------------------------------------------------
// Degree accumulation: deg_out[src[e]] += 1, deg_in[dst[e]] += 1
// ---------------------------------------------------------------------------
__global__ void deg_kernel(const int* __restrict__ src, const int* __restrict__ dst,
                           float* __restrict__ deg_out, float* __restrict__ deg_in, int E) {
    int e = blockIdx.x * blockDim.x + threadIdx.x;
    if (e < E) {
        atomicAdd(&deg_out[src[e]], 1.0f);
        atomicAdd(&deg_in[dst[e]], 1.0f);
    }
}

// In-place: deg -> rsqrt(max(deg,1))
__global__ void norm_kernel(float* __restrict__ deg_out, float* __restrict__ deg_in, int N) {
    int i = blockIdx.x * blockDim.x + threadIdx.x;
    if (i < N) {
        float d0 = deg_out[i];
        float d1 = deg_in[i];
        deg_out[i] = rsqrtf(d0 > 0.0f ? d0 : 1.0f);
        deg_in[i]  = rsqrtf(d1 > 0.0f ? d1 : 1.0f);
    }
}

// ---------------------------------------------------------------------------
// H[m, :COLS] = (X[m, :K] * norm[m]) @ W[K, COLS]   via V_WMMA_F32_16X16X4_F32
// One wave handles a 16-row strip across all COLS (COLS/16 accumulator tiles).
//
// fp32 A 16x4 layout: lane l -> row m = l&15; VGPR0 = K=2*(l>>4), VGPR1 = K+1.
// fp32 B 4x16 layout: lane l -> col n = l&15; VGPR v = K = v + 2*(l>>4).
// fp32 C/D 16x16:     lane l, VGPR v -> row = v + 8*(l>>4), col = l&15.
// ---------------------------------------------------------------------------
template <int K, int COLS>
__global__ void __launch_bounds__(128)
gemm_norm_wmma(const float* __restrict__ X, const float* __restrict__ normv,
               const float* __restrict__ W, float* __restrict__ H, int nrows) {
    constexpr int NT = COLS / 16;
    const int lane  = threadIdx.x & 31;
    const int wave  = threadIdx.x >> 5;
    const int tile_m = blockIdx.x * (blockDim.x >> 5) + wave;
    if (tile_m * 16 >= nrows) return;   // wave-uniform: EXEC stays all-1s

    const int mr = lane & 15;           // row within tile (A), col within tile (B/D)
    const int kh = (lane >> 4) << 1;    // K sub-offset: 0 or 2
    const int m  = tile_m * 16 + mr;

    const float nm = normv[m];
    const float* __restrict__ xrow = X + (long)m * K;

    v8f acc[NT];
#pragma unroll
    for (int t = 0; t < NT; ++t) { v8f z = {}; acc[t] = z; }

#pragma unroll 4
    for (int k = 0; k < K; k += 4) {
        v2f a;
        a.x = xrow[k + kh]     * nm;
        a.y = xrow[k + kh + 1] * nm;
        const float* __restrict__ wr = W + (long)(k + kh) * COLS;
#pragma unroll
        for (int t = 0; t < NT; ++t) {
            v2f b;
            b.x = wr[t * 16 + mr];
            b.y = wr[COLS + t * 16 + mr];
            acc[t] = __builtin_amdgcn_wmma_f32_16x16x4_f32(
                false, a, false, b, (short)0, acc[t], false, false);
        }
    }

    const int mbase = tile_m * 16 + ((lane >> 4) << 3);
#pragma unroll
    for (int t = 0; t < NT; ++t) {
#pragma unroll
        for (int v = 0; v < 8; ++v) {
            H[(long)(mbase + v) * COLS + t * 16 + mr] = acc[t][v];
        }
    }
}

// ---------------------------------------------------------------------------
// Edge aggregation: agg[dst[e], c] += H[src[e], c]   (C = 64 or 32, pow2)
// One thread per (edge, channel); channel-contiguous => coalesced.
// ---------------------------------------------------------------------------
template <int C>
__global__ void edge_agg(const float* __restrict__ H, const int* __restrict__ src,
                         const int* __restrict__ dst, float* __restrict__ agg, int E) {
    long idx = (long)blockIdx.x * blockDim.x + threadIdx.x;
    int e = (int)(idx / C);
    int c = (int)(idx & (C - 1));
    if (e >= E) return;
    float v = H[(long)src[e] * C + c];
    atomicAdd(&agg[(long)dst[e] * C + c], v);
}

// ---------------------------------------------------------------------------
// In-place finalize: x = agg * norm_in[row] + bias[col]; optional ReLU
// ---------------------------------------------------------------------------
template <int C, bool RELU>
__global__ void finalize_kernel(float* __restrict__ agg, const float* __restrict__ norm_in,
                                const float* __restrict__ bias, long total) {
    long idx = (long)blockIdx.x * blockDim.x + threadIdx.x;
    if (idx < total) {
        int row = (int)(idx / C);
        int col = (int)(idx & (C - 1));
        float v = agg[idx] * norm_in[row] + bias[col];
        agg[idx] = RELU ? (v > 0.0f ? v : 0.0f) : v;
    }
}

// ---------------------------------------------------------------------------
extern "C" void kernel_launch(void* const* d_in, const int* in_sizes, int n_in,
                              void* d_out, int out_size, void* d_ws, size_t ws_size,
                              hipStream_t stream) {
    const float* features = (const float*)d_in[0];
    const int*   src      = (const int*)d_in[1];
    const int*   dst      = (const int*)d_in[2];
    const float* W1       = (const float*)d_in[3];
    const float* b1       = (const float*)d_in[4];
    const float* W2       = (const float*)d_in[5];
    const float* b2       = (const float*)d_in[6];

    const int IN_F = 128, HID = 64, CLS = 32;
    const int N = in_sizes[0] / IN_F;   // 100000
    const int E = in_sizes[1];          // 1600000

    // Workspace layout (floats): norms (2N) | h buffer (64N) | agg/x1 (64N)
    float* ws       = (float*)d_ws;
    float* norm_out = ws;                       // N  (deg_out -> norm_out in place)
    float* norm_in  = ws + (long)N;             // N  (deg_in  -> norm_in  in place)
    float* hbuf     = ws + 2L * N;              // N*64 (h1; reused for h2's N*32)
    float* aggbuf   = ws + 2L * N + 64L * N;    // N*64 (agg1 -> x1 in place)
    float* outp     = (float*)d_out;            // N*32 (agg2 -> output in place)

    // Zero the accumulators (graph-capturable async memsets).
    hipMemsetAsync(norm_out, 0, sizeof(float) * 2L * N, stream);
    hipMemsetAsync(aggbuf,   0, sizeof(float) * 64L * N, stream);
    hipMemsetAsync(outp,     0, sizeof(float) * (size_t)out_size, stream);

    // Degrees -> normalization factors.
    deg_kernel<<<(E + 255) / 256, 256, 0, stream>>>(src, dst, norm_out, norm_in, E);
    norm_kernel<<<(N + 255) / 256, 256, 0, stream>>>(norm_out, norm_in, N);

    const int mtiles = (N + 15) / 16;            // 6250 row tiles
    const int gblocks = (mtiles + 3) / 4;        // 4 waves / block

    // ---- Layer 1 ----
    gemm_norm_wmma<128, 64><<<gblocks, 128, 0, stream>>>(features, norm_out, W1, hbuf, N);
    {
        long tot = (long)E * HID;
        edge_agg<64><<<(unsigned)((tot + 255) / 256), 256, 0, stream>>>(hbuf, src, dst, aggbuf, E);
        long n1 = (long)N * HID;
        finalize_kernel<64, true><<<(unsigned)((n1 + 255) / 256), 256, 0, stream>>>(
            aggbuf, norm_in, b1, n1);
    }

    // ---- Layer 2 ----
    gemm_norm_wmma<64, 32><<<gblocks, 128, 0, stream>>>(aggbuf, norm_out, W2, hbuf, N);
    {
        long tot = (long)E * CLS;
        edge_agg<32><<<(unsigned)((tot + 255) / 256), 256, 0, stream>>>(hbuf, src, dst, outp, E);
        long n2 = (long)N * CLS;
        finalize_kernel<32, false><<<(unsigned)((n2 + 255) / 256), 256, 0, stream>>>(
            outp, norm_in, b2, n2);
    }
}